// VisualAttentionProcess_24618752540874
// MI455X (gfx1250) — compile-verified
//
#include <hip/hip_runtime.h>
#include <math.h>

typedef float v2f __attribute__((ext_vector_type(2)));
typedef float v8f __attribute__((ext_vector_type(8)));
typedef int   v4i __attribute__((vector_size(4 * sizeof(int))));

#define AS1 __attribute__((address_space(1)))
#define AS3 __attribute__((address_space(3)))

#define N_HEADS   8
#define HEAD_DIM  160
#define L_TXT     77
#define SIG_A     10.0f
#define SIG_B     0.5f
#define SIG_C     1.0f
#define EPSV      1e-8f

// ---------------------------------------------------------------------------
// CDNA5 async copy global -> LDS (ASYNCcnt-tracked), 16B per lane.
// ---------------------------------------------------------------------------
__device__ __forceinline__ void async_b128(void* dst_lds, const void* src_glob) {
#if __has_builtin(__builtin_amdgcn_global_load_async_to_lds_b128)
    __builtin_amdgcn_global_load_async_to_lds_b128(
        (AS1 v4i*)src_glob, (AS3 v4i*)dst_lds, 0, 0);
#else
    asm volatile("global_load_async_to_lds_b128 %0, %1, off"
                 :: "v"((AS3 v4i*)dst_lds), "v"((AS1 v4i*)src_glob)
                 : "memory");
#endif
}

__device__ __forceinline__ void wait_async0() {
#if __has_builtin(__builtin_amdgcn_s_wait_asynccnt)
    __builtin_amdgcn_s_wait_asynccnt(0);
#else
    asm volatile("s_wait_asynccnt 0x0" ::: "memory");
#endif
}

// ---------------------------------------------------------------------------
// GEMM: C[M,N] = A[M,K] * B[K,N] (+bias), fp32 via V_WMMA_F32_16X16X4_F32.
// Block = 4 waves (128 threads) -> 64x64 tile, K blocked by 16 with
// double-buffered async-to-LDS staging of both A and B tiles.
// N % 64 == 0, K % 16 == 0; M row-guarded (OOB rows clamped: a WMMA D row
// depends only on its own A row, so garbage rows never reach stored output).
// ---------------------------------------------------------------------------
#define BM 64
#define BN 64
#define BK 16
#define APAD 20   // 80B rows: 16B-aligned, bank-conflict-free

__global__ __launch_bounds__(128)
void gemm_f32_wmma(const float* __restrict__ A, const float* __restrict__ B,
                   const float* __restrict__ bias, float* __restrict__ C,
                   int M, int N, int K) {
    __shared__ float As[2][BM][APAD];   // A tile, row-major [m][k]
    __shared__ float Bs[2][BK][BN];     // B tile, row-major [k][n]

    const int tid   = threadIdx.x;
    const int lane  = tid & 31;
    const int wave  = tid >> 5;
    const int l16   = lane & 15;
    const int khalf = (lane >> 4) << 1;          // 0 or 2
    const int m0    = blockIdx.y * BM;
    const int n0    = blockIdx.x * BN;

    auto issue = [&](int buf, int k0) {
        #pragma unroll
        for (int i = 0; i < 2; ++i) {
            const int c  = tid + i * 128;        // 0..255
            const int r  = c >> 2, q = c & 3;    // A: row, quad
            int grow = m0 + r; if (grow >= M) grow = M - 1;   // clamp, no zeroing
            async_b128(&As[buf][r][q * 4],
                       A + (size_t)grow * (size_t)K + k0 + q * 4);
            const int kr = c >> 4, f4 = c & 15;  // B: k-row, float4
            async_b128(&Bs[buf][kr][f4 * 4],
                       B + (size_t)(k0 + kr) * (size_t)N + n0 + f4 * 4);
        }
    };

    v8f acc[4];
    #pragma unroll
    for (int t = 0; t < 4; ++t) acc[t] = (v8f){0.f,0.f,0.f,0.f,0.f,0.f,0.f,0.f};

    const int KT = K / BK;
    issue(0, 0);
    const int arow = wave * 16 + l16;
    for (int kt = 0; kt < KT; ++kt) {
        const int buf = kt & 1;
        wait_async0();
        __syncthreads();
        if (kt + 1 < KT) issue(buf ^ 1, (kt + 1) * BK);
        #pragma unroll
        for (int kk = 0; kk < 4; ++kk) {
            v2f a;
            a.x = As[buf][arow][kk * 4 + khalf];
            a.y = As[buf][arow][kk * 4 + khalf + 1];
            #pragma unroll
            for (int t = 0; t < 4; ++t) {
                v2f b;
                b.x = Bs[buf][kk * 4 + khalf][t * 16 + l16];
                b.y = Bs[buf][kk * 4 + khalf + 1][t * 16 + l16];
                acc[t] = __builtin_amdgcn_wmma_f32_16x16x4_f32(
                    false, a, false, b, (short)0, acc[t], false, false);
            }
        }
    }

    const int rbase = m0 + wave * 16 + ((lane >> 4) << 3);
    #pragma unroll
    for (int t = 0; t < 4; ++t) {
        const int col = n0 + t * 16 + l16;
        const float bv = bias ? bias[col] : 0.0f;
        #pragma unroll
        for (int i = 0; i < 8; ++i) {
            const int row = rbase + i;
            if (row < M) C[(size_t)row * (size_t)N + col] = acc[t][i] + bv;
        }
    }
}

// ---------------------------------------------------------------------------
// Concept-target element: target_value is [2*C*8, 77, 160]; regrouped index
// (half, c, h) -> row half*24 + c*8 + h.
// ---------------------------------------------------------------------------
__device__ __forceinline__ float tv_elem(const float* __restrict__ tv,
                                         int l, int c, int half, int h, int dd) {
    return tv[((size_t)((half * 24 + c * 8 + h) * L_TXT + l)) * HEAD_DIM + dd];
}

// ---------------------------------------------------------------------------
// Per-token prep: Gram G (3x3), Pm = I - tril(G/diag, -1),
// dot2[c] = Pm[c]·G·Pm[c]^T, nT[c] = max(sqrt(G[c][c]), EPS).
// params stride 16 per token: [0..8]=Pm, [9..11]=dot2, [12..14]=nT.
// ---------------------------------------------------------------------------
__global__ __launch_bounds__(256)
void prep_tokens(const float* __restrict__ tv, float* __restrict__ params) {
    const int l   = blockIdx.x;
    const int tid = threadIdx.x;
    float g[6] = {0.f, 0.f, 0.f, 0.f, 0.f, 0.f};
    for (int idx = tid; idx < 2560; idx += 256) {
        const int half = (idx >= 1280) ? 1 : 0;
        const int rem  = idx - half * 1280;
        const int h    = rem / HEAD_DIM;
        const int dd   = rem - h * HEAD_DIM;
        const float t0 = tv_elem(tv, l, 0, half, h, dd);
        const float t1 = tv_elem(tv, l, 1, half, h, dd);
        const float t2 = tv_elem(tv, l, 2, half, h, dd);
        g[0] += t0 * t0; g[1] += t0 * t1; g[2] += t0 * t2;
        g[3] += t1 * t1; g[4] += t1 * t2; g[5] += t2 * t2;
    }
    __shared__ float red[6][256];
    for (int j = 0; j < 6; ++j) red[j][tid] = g[j];
    __syncthreads();
    for (int s = 128; s > 0; s >>= 1) {
        if (tid < s)
            for (int j = 0; j < 6; ++j) red[j][tid] += red[j][tid + s];
        __syncthreads();
    }
    if (tid == 0) {
        float G[3][3];
        G[0][0] = red[0][0]; G[0][1] = G[1][0] = red[1][0];
        G[0][2] = G[2][0] = red[2][0];
        G[1][1] = red[3][0]; G[1][2] = G[2][1] = red[4][0];
        G[2][2] = red[5][0];
        float Pm[3][3];
        for (int c = 0; c < 3; ++c)
            for (int e = 0; e < 3; ++e)
                Pm[c][e] = ((c == e) ? 1.0f : 0.0f)
                         - ((c > e) ? (G[c][e] / G[e][e]) : 0.0f);
        float* p = params + (size_t)l * 16;
        for (int c = 0; c < 3; ++c)
            for (int e = 0; e < 3; ++e) p[c * 3 + e] = Pm[c][e];
        for (int c = 0; c < 3; ++c) {
            float d2 = 0.0f;
            for (int e = 0; e < 3; ++e)
                for (int f = 0; f < 3; ++f)
                    d2 += Pm[c][e] * G[e][f] * Pm[c][f];
            p[9 + c] = d2;
            float nt = sqrtf(G[c][c]);
            p[12 + c] = (nt < EPSV) ? EPSV : nt;
        }
    }
}

// ---------------------------------------------------------------------------
// Retain kernel: one block per (record r, token l).
// ---------------------------------------------------------------------------
__global__ __launch_bounds__(256)
void retain_kernel(const float* __restrict__ tv, const float* __restrict__ vbuf,
                   const float* __restrict__ params, float* __restrict__ vret) {
    const int r   = blockIdx.x;
    const int l   = blockIdx.y;
    const int tid = threadIdx.x;

    float td[3] = {0.f, 0.f, 0.f};
    float np2 = 0.f;
    for (int idx = tid; idx < 2560; idx += 256) {
        const int half = (idx >= 1280) ? 1 : 0;
        const int rem  = idx - half * 1280;
        const int h    = rem / HEAD_DIM;
        const int dd   = rem - h * HEAD_DIM;
        const int b    = half * 4 + r;
        const float pv = vbuf[((size_t)(b * L_TXT + l)) * 1280 + h * HEAD_DIM + dd];
        td[0] += tv_elem(tv, l, 0, half, h, dd) * pv;
        td[1] += tv_elem(tv, l, 1, half, h, dd) * pv;
        td[2] += tv_elem(tv, l, 2, half, h, dd) * pv;
        np2   += pv * pv;
    }
    __shared__ float red[4][256];
    red[0][tid] = td[0]; red[1][tid] = td[1]; red[2][tid] = td[2]; red[3][tid] = np2;
    __syncthreads();
    for (int s = 128; s > 0; s >>= 1) {
        if (tid < s)
            for (int j = 0; j < 4; ++j) red[j][tid] += red[j][tid + s];
        __syncthreads();
    }
    __shared__ float coef[3];
    if (tid == 0) {
        const float* p = params + (size_t)l * 16;
        float Pm[3][3];
        for (int c = 0; c < 3; ++c)
            for (int e = 0; e < 3; ++e) Pm[c][e] = p[c * 3 + e];
        float tdr[3] = {red[0][0], red[1][0], red[2][0]};
        float w[3];
        for (int c = 0; c < 3; ++c) {
            float d1 = Pm[c][0] * tdr[0] + Pm[c][1] * tdr[1] + Pm[c][2] * tdr[2];
            float wv = d1 / p[9 + c];
            if (wv != wv) wv = 0.0f;                       // nan_to_num
            else if (wv >  3.4028235e38f) wv =  3.4028235e38f;
            else if (wv < -3.4028235e38f) wv = -3.4028235e38f;
            if (l == 0) wv = 0.0f;                         // zero BOS token
            w[c] = wv;
        }
        float nP = sqrtf(red[3][0]);
        if (nP < EPSV) nP = EPSV;
        float cs[3];
        for (int c = 0; c < 3; ++c) {
            float cv = tdr[c] / (p[12 + c] * nP);
            cs[c] = SIG_C / (1.0f + expf(-SIG_A * (cv - SIG_B)));
        }
        for (int e = 0; e < 3; ++e)
            coef[e] = cs[e] * (w[0] * Pm[0][e] + w[1] * Pm[1][e] + w[2] * Pm[2][e]);
    }
    __syncthreads();

    for (int idx = tid; idx < 2560; idx += 256) {
        const int half = (idx >= 1280) ? 1 : 0;
        const int rem  = idx - half * 1280;
        const int h    = rem / HEAD_DIM;
        const int dd   = rem - h * HEAD_DIM;
        const int b    = half * 4 + r;
        const size_t off = ((size_t)(b * L_TXT + l)) * 1280 + h * HEAD_DIM + dd;
        const float pv = vbuf[off];
        const float era = coef[0] * tv_elem(tv, l, 0, half, h, dd)
                        + coef[1] * tv_elem(tv, l, 1, half, h, dd)
                        + coef[2] * tv_elem(tv, l, 2, half, h, dd);
        vret[off] = pv - era;
    }
}

// ---------------------------------------------------------------------------
// WMMA attention.  grid = (S/64, B*HEADS), block = 128 (4 waves x 16 queries).
// Phase 1: S^T[key][query] = K·Q^T  (K async-staged in LDS [key][dd], rows
//   padded 77->80 with zeros, row pitch 164 floats for conflict-free A-frag
//   reads; Q^T read from global as per-lane b64).
// Softmax: in S^T's D-layout, lane = query (lanes L and L+16 hold
//   complementary keys of query L), so row max/sum = 40-reg local reduction
//   + one __shfl_xor(16).  Pad keys masked to -inf -> exp == 0.
// Phase 2: each lane writes its query's P row to LDS [query][key] (pitch 84),
//   which IS the A-fragment layout for O = P·V; V async-staged [key][dd].
// Dynamic LDS (122 KB < 320 KB WGP budget; avoids static 64 KB cap).
// ---------------------------------------------------------------------------
#define KPITCH 164   // l16*164 mod 64 -> 16 distinct banks
#define PPITCH 84    // l16*84  mod 64 -> 16 distinct banks
#define ATT_LDS_FLOATS (80 * KPITCH + 80 * HEAD_DIM + 4 * 16 * PPITCH)

__global__ __launch_bounds__(128)
void attn_wmma_kernel(const float* __restrict__ qb, const float* __restrict__ kb,
                      const float* __restrict__ vb, float* __restrict__ ob) {
    extern __shared__ float smem[];
    float (*Ks)[KPITCH] = reinterpret_cast<float(*)[KPITCH]>(smem);
    float (*Vs)[HEAD_DIM] = reinterpret_cast<float(*)[HEAD_DIM]>(smem + 80 * KPITCH);

    const int tid  = threadIdx.x;
    const int lane = tid & 31;
    const int wave = tid >> 5;
    const int l16  = lane & 15;
    const int half = lane >> 4;
    const int khalf = half << 1;

    float (*Ps)[PPITCH] = reinterpret_cast<float(*)[PPITCH]>(
        smem + 80 * KPITCH + 80 * HEAD_DIM + wave * 16 * PPITCH);

    const int bh = blockIdx.y;
    const int b  = bh >> 3;
    const int h  = bh & 7;
    const int q0 = blockIdx.x * 64 + wave * 16;

    // zero pad rows 77..79
    for (int c = tid; c < 3 * HEAD_DIM; c += 128) {
        Ks[77 + c / HEAD_DIM][c % HEAD_DIM] = 0.0f;
        Vs[77 + c / HEAD_DIM][c % HEAD_DIM] = 0.0f;
    }
    // async-stage K and V tiles: 77 rows x 40 float4 each
    for (int c = tid; c < 77 * 40; c += 128) {
        const int row = c / 40, q4 = (c % 40) * 4;
        const size_t src = (size_t)(b * L_TXT + row) * 1280 + h * HEAD_DIM + q4;
        async_b128(&Ks[row][q4], kb + src);
        async_b128(&Vs[row][q4], vb + src);
    }
    wait_async0();
    __syncthreads();

    // ---- scores: S^T = K · Q^T ----
    v8f s[5];
    #pragma unroll
    for (int t = 0; t < 5; ++t) s[t] = (v8f){0.f,0.f,0.f,0.f,0.f,0.f,0.f,0.f};
    const float* qrow = qb + (size_t)(b * 1024 + q0 + l16) * 1280 + h * HEAD_DIM;
    for (int d0 = 0; d0 < HEAD_DIM; d0 += 4) {
        v2f bq;
        bq.x = qrow[d0 + khalf];
        bq.y = qrow[d0 + khalf + 1];
        #pragma unroll
        for (int mt = 0; mt < 5; ++mt) {
            v2f ak;
            ak.x = Ks[mt * 16 + l16][d0 + khalf];
            ak.y = Ks[mt * 16 + l16][d0 + khalf + 1];
            s[mt] = __builtin_amdgcn_wmma_f32_16x16x4_f32(
                false, ak, false, bq, (short)0, s[mt], false, false);
        }
    }

    // ---- softmax over keys (per lane: query = l16) ----
    const float scale = 1.0f / sqrtf((float)HEAD_DIM);
    #pragma unroll
    for (int mt = 0; mt < 5; ++mt)
        #pragma unroll
        for (int i = 0; i < 8; ++i) {
            float v = s[mt][i] * scale;
            // key = mt*16 + half*8 + i; pads (>=77) masked to -inf
            if (mt == 4 && (half * 8 + i) >= 13) v = -INFINITY;
            s[mt][i] = v;
        }
    float mx = -INFINITY;
    #pragma unroll
    for (int mt = 0; mt < 5; ++mt)
        #pragma unroll
        for (int i = 0; i < 8; ++i) mx = fmaxf(mx, s[mt][i]);
    mx = fmaxf(mx, __shfl_xor(mx, 16, 32));
    float sum = 0.0f;
    #pragma unroll
    for (int mt = 0; mt < 5; ++mt)
        #pragma unroll
        for (int i = 0; i < 8; ++i) {
            float e = expf(s[mt][i] - mx);
            s[mt][i] = e;
            sum += e;
        }
    sum += __shfl_xor(sum, 16, 32);
    const float inv = 1.0f / sum;

    // write P row (query = l16) in [query][key] layout == A-fragment layout
    #pragma unroll
    for (int mt = 0; mt < 5; ++mt)
        #pragma unroll
        for (int i = 0; i < 8; ++i)
            Ps[l16][mt * 16 + half * 8 + i] = s[mt][i] * inv;
    // Ps is wave-private; same-wave LDS RAW ordered by DScnt (compiler waits)

    // ---- output: O = P · V ----
    v8f o[10];
    #pragma unroll
    for (int t = 0; t < 10; ++t) o[t] = (v8f){0.f,0.f,0.f,0.f,0.f,0.f,0.f,0.f};
    for (int k0 = 0; k0 < 80; k0 += 4) {
        v2f ap;
        ap.x = Ps[l16][k0 + khalf];
        ap.y = Ps[l16][k0 + khalf + 1];
        #pragma unroll
        for (int t = 0; t < 10; ++t) {
            v2f bv;
            bv.x = Vs[k0 + khalf][t * 16 + l16];
            bv.y = Vs[k0 + khalf + 1][t * 16 + l16];
            o[t] = __builtin_amdgcn_wmma_f32_16x16x4_f32(
                false, ap, false, bv, (short)0, o[t], false, false);
        }
    }
    #pragma unroll
    for (int t = 0; t < 10; ++t)
        #pragma unroll
        for (int i = 0; i < 8; ++i)
            ob[(size_t)(b * 1024 + q0 + half * 8 + i) * 1280
               + h * HEAD_DIM + t * 16 + l16] = o[t][i];
}

// ---------------------------------------------------------------------------
extern "C" void kernel_launch(void* const* d_in, const int* in_sizes, int n_in,
                              void* d_out, int out_size, void* d_ws, size_t ws_size,
                              hipStream_t stream) {
    (void)in_sizes; (void)n_in; (void)out_size; (void)ws_size;
    const float* hidden = (const float*)d_in[0];   // [8,1024,1280]
    const float* enc    = (const float*)d_in[1];   // [8,77,768]
    const float* Wq     = (const float*)d_in[2];   // [1280,1280]
    const float* Wk     = (const float*)d_in[3];   // [768,1280]
    const float* Wv     = (const float*)d_in[4];   // [768,1280]
    const float* Wo     = (const float*)d_in[5];   // [1280,1280]
    const float* bo     = (const float*)d_in[6];   // [1280]
    const float* tv     = (const float*)d_in[7];   // [48,77,160]
    float* out = (float*)d_out;

    const size_t MQ = 8192, MKV = 616, DD = 1280;
    float* ws    = (float*)d_ws;
    float* qbuf  = ws;               ws += MQ  * DD;
    float* kbuf  = ws;               ws += MKV * DD;
    float* vbuf  = ws;               ws += MKV * DD;
    float* vret  = ws;               ws += MKV * DD;
    float* attn  = ws;               ws += MQ  * DD;
    float* prm   = ws;               ws += L_TXT * 16;

    gemm_f32_wmma<<<dim3(1280 / 64, 8192 / 64), 128, 0, stream>>>(
        hidden, Wq, nullptr, qbuf, 8192, 1280, 1280);
    gemm_f32_wmma<<<dim3(1280 / 64, (616 + 63) / 64), 128, 0, stream>>>(
        enc, Wk, nullptr, kbuf, 616, 1280, 768);
    gemm_f32_wmma<<<dim3(1280 / 64, (616 + 63) / 64), 128, 0, stream>>>(
        enc, Wv, nullptr, vbuf, 616, 1280, 768);
    prep_tokens<<<L_TXT, 256, 0, stream>>>(tv, prm);
    retain_kernel<<<dim3(4, L_TXT), 256, 0, stream>>>(tv, vbuf, prm, vret);
    attn_wmma_kernel<<<dim3(1024 / 64, 64), 128,
                       ATT_LDS_FLOATS * sizeof(float), stream>>>(
        qbuf, kbuf, vret, attn);
    gemm_f32_wmma<<<dim3(1280 / 64, 8192 / 64), 128, 0, stream>>>(
        attn, Wo, bo, out, 8192, 1280, 1280);
}